// E64AdditiveHCell_4887672783100
// MI455X (gfx1250) — compile-verified
//
#include <hip/hip_runtime.h>
#include <cstdint>
#include <cstddef>

// ---------------------------------------------------------------------------
// Problem constants (from reference): T=2048, B=8, D=1024
// ---------------------------------------------------------------------------
#define T_STEPS 2048
#define BATCH   8
#define DIM     1024
#define M_TOT   (T_STEPS * BATCH)   // 16384 rows of X
#define K_TOT   DIM                 // 1024
#define N_TOT   DIM                 // 1024

// GEMM tiling
#define BM 128        // rows per block (8 waves x 16)
#define BN 64         // cols per block (each wave covers all 64)
#define BK 64         // K chunk staged in LDS
#define PAD_A 68      // sA row stride in floats (272B: 16B-aligned, bank-skewed)

typedef __attribute__((ext_vector_type(2))) float v2f;
typedef __attribute__((ext_vector_type(4))) float v4f;
typedef __attribute__((ext_vector_type(8))) float v8f;

__device__ __forceinline__ float sigmoid_f(float x) {
    return 1.0f / (1.0f + __expf(-x));
}

// ---------------------------------------------------------------------------
// Kernel 1: fused dual GEMM, software-pipelined across K chunks.
//   alpha_out[m,n] = sigmoid( sum_k X[m,k]*Wa[n,k] + ba[n] )
//   wx_out[m,n]    =          sum_k X[m,k]*Wx[n,k] + bb[n]
// V_WMMA_F32_16X16X4_F32 (fp32 matrix path, full precision vs reference).
// Global loads for chunk k+1 are issued through registers before the
// 128-WMMA compute block of chunk k, hiding HBM/L2 latency.
// ---------------------------------------------------------------------------
__global__ __launch_bounds__(256)
void e64_fused_proj_gemm(const float* __restrict__ X,
                         const float* __restrict__ Wa,
                         const float* __restrict__ ba,
                         const float* __restrict__ Wx,
                         const float* __restrict__ bb,
                         float* __restrict__ alpha_out,
                         float* __restrict__ wx_out)
{
    __shared__ float sA[BM * PAD_A];          // 128 x 64 A tile (padded rows)
    __shared__ v2f   sBa[(BK / 2) * BN];      // sB[p][n] = {B[2p][n], B[2p+1][n]}
    __shared__ v2f   sBx[(BK / 2) * BN];

    const int tid   = threadIdx.x;            // 0..255, 8 waves
    const int wave  = tid >> 5;               // 0..7
    const int lane  = tid & 31;
    const int lhalf = lane >> 4;              // 0 or 1 (lane half)
    const int l16   = lane & 15;

    const int n0 = blockIdx.x * BN;
    const int m0 = blockIdx.y * BM;

    v8f acc_a[4];
    v8f acc_x[4];
#pragma unroll
    for (int j = 0; j < 4; ++j) { acc_a[j] = (v8f)0.0f; acc_x[j] = (v8f)0.0f; }

    const int mrow = (wave << 4) + l16;       // this lane's A row within the block tile

    // Per-thread staging coordinates (fixed across chunks)
    // A: 8 float4 per thread
    // B: 4 float4 per thread per weight
    v4f rA[8];
    v4f rBa[4];
    v4f rBx[4];

    // ---- prologue: prefetch chunk 0 into registers ----
#pragma unroll
    for (int i = 0; i < 8; ++i) {
        int idx = tid + i * 256;
        int row = idx >> 4;                   // 0..127
        int c4  = (idx & 15) << 2;            // 0..60 step 4
        rA[i] = *reinterpret_cast<const v4f*>(X + (size_t)(m0 + row) * K_TOT + c4);
    }
#pragma unroll
    for (int i = 0; i < 4; ++i) {
        int idx = tid + i * 256;
        int n   = idx & 63;                   // consecutive tids -> consecutive n
        int k4  = (idx >> 6) << 2;            // 0..60 step 4
        rBa[i] = *reinterpret_cast<const v4f*>(Wa + (size_t)(n0 + n) * K_TOT + k4);
        rBx[i] = *reinterpret_cast<const v4f*>(Wx + (size_t)(n0 + n) * K_TOT + k4);
    }

    for (int k0 = 0; k0 < K_TOT; k0 += BK) {
        // Ensure previous chunk's compute is done reading LDS.
        __syncthreads();

        // ---- stage registers -> LDS ----
#pragma unroll
        for (int i = 0; i < 8; ++i) {
            int idx = tid + i * 256;
            int row = idx >> 4;
            int c4  = (idx & 15) << 2;
            *reinterpret_cast<v4f*>(&sA[row * PAD_A + c4]) = rA[i];
        }
#pragma unroll
        for (int i = 0; i < 4; ++i) {
            int idx = tid + i * 256;
            int n   = idx & 63;
            int k4  = (idx >> 6) << 2;
            int p   = k4 >> 1;
            v2f t0; t0.x = rBa[i].x; t0.y = rBa[i].y;
            v2f t1; t1.x = rBa[i].z; t1.y = rBa[i].w;
            v2f t2; t2.x = rBx[i].x; t2.y = rBx[i].y;
            v2f t3; t3.x = rBx[i].z; t3.y = rBx[i].w;
            sBa[p * BN + n]       = t0;
            sBa[(p + 1) * BN + n] = t1;
            sBx[p * BN + n]       = t2;
            sBx[(p + 1) * BN + n] = t3;
        }
        __syncthreads();

        // ---- issue next chunk's global loads (overlap with compute below) ----
        const int kn = k0 + BK;
        if (kn < K_TOT) {
#pragma unroll
            for (int i = 0; i < 8; ++i) {
                int idx = tid + i * 256;
                int row = idx >> 4;
                int c4  = (idx & 15) << 2;
                rA[i] = *reinterpret_cast<const v4f*>(X + (size_t)(m0 + row) * K_TOT + kn + c4);
            }
#pragma unroll
            for (int i = 0; i < 4; ++i) {
                int idx = tid + i * 256;
                int n   = idx & 63;
                int k4  = (idx >> 6) << 2;
                rBa[i] = *reinterpret_cast<const v4f*>(Wa + (size_t)(n0 + n) * K_TOT + kn + k4);
                rBx[i] = *reinterpret_cast<const v4f*>(Wx + (size_t)(n0 + n) * K_TOT + kn + k4);
            }
        }

        // ---- compute: per 4-wide k-step, 1 A fragment feeds 8 WMMAs ----
#pragma unroll
        for (int ks = 0; ks < BK; ks += 4) {
            // A 16x4 fragment: lanes 0-15 -> K = ks,ks+1 ; lanes 16-31 -> K = ks+2,ks+3
            v2f a = *reinterpret_cast<const v2f*>(&sA[mrow * PAD_A + ks + (lhalf << 1)]);
            int p = (ks >> 1) + lhalf;        // K-pair index for this lane half
#pragma unroll
            for (int j = 0; j < 4; ++j) {
                v2f bA = sBa[p * BN + (j << 4) + l16];
                v2f bX = sBx[p * BN + (j << 4) + l16];
                acc_a[j] = __builtin_amdgcn_wmma_f32_16x16x4_f32(
                    false, a, false, bA, (short)0, acc_a[j], false, false);
                acc_x[j] = __builtin_amdgcn_wmma_f32_16x16x4_f32(
                    false, a, false, bX, (short)0, acc_x[j], false, false);
            }
        }
    }

    // ---- epilogue: C layout -> VGPR i holds row (i + 8*lane_half), col = lane&15 ----
    const int mbase = m0 + (wave << 4) + (lhalf << 3);
#pragma unroll
    for (int j = 0; j < 4; ++j) {
        int n = n0 + (j << 4) + l16;
        float biasA = ba[n];
        float biasX = bb[n];
#pragma unroll
        for (int i = 0; i < 8; ++i) {
            size_t off = (size_t)(mbase + i) * N_TOT + n;
            alpha_out[off] = sigmoid_f(acc_a[j][i] + biasA);
            wx_out[off]    = acc_x[j][i] + biasX;
        }
    }
}

// ---------------------------------------------------------------------------
// Kernel 2: elementwise-parallel scan over T. One thread per (b,d) chain.
// alpha_all / wx_all / out / h_out intentionally NOT __restrict__: they may
// alias (workspace-fallback mode). Per-thread program order guarantees each
// slot t is read (incl. the t+1 pipeline load) before it is overwritten.
// ---------------------------------------------------------------------------
__global__ __launch_bounds__(128)
void e64_scan_kernel(const float* alpha_all,
                     const float* wx_all,
                     const float* __restrict__ h0,
                     float* out,
                     float* h_out)
{
    const int S  = BATCH * DIM;                         // 8192 chains
    const int id = blockIdx.x * blockDim.x + threadIdx.x;

    // Load t=0 inputs BEFORE writing h_out[0] (required in aliasing mode).
    float a  = alpha_all[id];
    float wx = wx_all[id];
    float h  = h0[id];
    h_out[id] = h;

    for (int t = 0; t < T_STEPS; ++t) {
        const int tn = t + 1;
        float a_n = 0.0f, wx_n = 0.0f;
        if (tn < T_STEPS) {                             // pipeline next step's inputs
            a_n  = alpha_all[(size_t)tn * S + id];
            wx_n = wx_all[(size_t)tn * S + id];
        }
        int tp = (t + 8 < T_STEPS) ? (t + 8) : (T_STEPS - 1);
        __builtin_prefetch(&alpha_all[(size_t)tp * S + id], 0, 0);
        __builtin_prefetch(&wx_all[(size_t)tp * S + id], 0, 0);

        float v = tanhf(h + wx);
        h = a * h + (1.0f - a) * v;
        float s = sigmoid_f(h);                         // silu(h) = h * s ; out = h * silu(h)
        out[(size_t)t * S + id]    = h * h * s;
        h_out[(size_t)tn * S + id] = h;

        a = a_n; wx = wx_n;
    }
}

// ---------------------------------------------------------------------------
// Launch
// ---------------------------------------------------------------------------
extern "C" void kernel_launch(void* const* d_in, const int* in_sizes, int n_in,
                              void* d_out, int out_size, void* d_ws, size_t ws_size,
                              hipStream_t stream)
{
    const float* x  = (const float*)d_in[0];   // [T,B,D] == [M,K]
    const float* h0 = (const float*)d_in[1];   // [B,D]
    const float* Wa = (const float*)d_in[2];   // [D,D] row-major [n][k]
    const float* ba = (const float*)d_in[3];   // [D]
    const float* Wx = (const float*)d_in[4];   // [D,D]
    const float* bb = (const float*)d_in[5];   // [D]

    float* out   = (float*)d_out;                      // [T,B,D]
    float* h_out = out + (size_t)M_TOT * N_TOT;        // [T+1,B,D]

    const size_t need = 2ull * (size_t)M_TOT * N_TOT * sizeof(float);  // 128 MB
    float* alpha_all;
    float* wx_all;
    if (ws_size >= need) {
        alpha_all = (float*)d_ws;
        wx_all    = alpha_all + (size_t)M_TOT * N_TOT;
    } else {
        // Fallback: stage projections in the output buffers; the scan consumes
        // slot t strictly before overwriting it (same thread, program order).
        alpha_all = out;
        wx_all    = h_out;
    }

    dim3 gemm_grid(N_TOT / BN, M_TOT / BM);            // 16 x 128 blocks
    e64_fused_proj_gemm<<<gemm_grid, 256, 0, stream>>>(x, Wa, ba, Wx, bb,
                                                       alpha_all, wx_all);

    e64_scan_kernel<<<(BATCH * DIM) / 128, 128, 0, stream>>>(alpha_all, wx_all,
                                                             h0, out, h_out);
}